// LengthRegulator_50165218017414
// MI455X (gfx1250) — compile-verified
//
#include <hip/hip_runtime.h>
#include <stdint.h>

// Problem constants (match reference).
#define B_  32
#define T_  512
#define D_  512
#define ML_ 4096            // MAX_LEN
#define ROWS_PER_BLOCK 8    // 4096 % 8 == 0 -> a block never straddles a batch

typedef __attribute__((ext_vector_type(4))) float v4f;  // native clang vector

// ---------------------------------------------------------------------------
// Expansion kernel (defined FIRST so the disasm snippet shows the async ops).
// One wave per block; each wave streams 8 output rows (2 KB each) through LDS
// with the CDNA5 async DMA path:
//   4x global_load_async_to_lds_b128  -> s_wait_asynccnt 0 ->
//   4x global_store_async_from_lds_b128 (th:TH_STORE_NT, streamed output)
// Double-buffered so row r loads overlap row r-1 stores. Masked rows (idx<0,
// always a contiguous tail within a block) write zeros with NT b128 stores.
// ---------------------------------------------------------------------------
__global__ __launch_bounds__(32) void lr_expand_kernel(const float* __restrict__ x,
                                                       const int* __restrict__ idxbuf,
                                                       float* __restrict__ out) {
  __shared__ alignas(16) float buf[2][D_];   // 2 x 2KB double buffer
  const int lane = threadIdx.x;              // 0..31 (wave32)
  const int row0 = blockIdx.x * ROWS_PER_BLOCK;

  // Generic LDS pointer: low 32 bits are the LDS byte offset (SHARED aperture
  // lives in the high bits) — exactly what the async ops' VDST/VSRC VGPR holds.
  const unsigned lds_base = (unsigned)(uintptr_t)(void*)&buf[0][0];

#pragma unroll 1
  for (int r = 0; r < ROWS_PER_BLOCK; ++r) {
    const int row = row0 + r;                // global output row (b*4096 + f)
    const int b   = row >> 12;               // 4096 frames per batch
    const int i   = idxbuf[row];             // wave-uniform source row or -1
    float* dst = out + (size_t)row * D_;

    if (i >= 0) {
      const float* src = x + ((size_t)(b * T_ + i)) * D_;
      const unsigned lds = lds_base + (unsigned)((r & 1) * (D_ * 4)) + (unsigned)(lane * 16);
      const unsigned long long ga = (unsigned long long)(uintptr_t)src + (unsigned)(lane * 16);

      // 32 lanes x 16 B x 4 issues = 2048 B row, memory -> LDS (ASYNCcnt).
      // x is L2-resident (32 MB << 192 MB L2), default RT load hint.
      asm volatile(
          "global_load_async_to_lds_b128 %0, %1, off\n\t"
          "global_load_async_to_lds_b128 %0, %1, off offset:512\n\t"
          "global_load_async_to_lds_b128 %0, %1, off offset:1024\n\t"
          "global_load_async_to_lds_b128 %0, %1, off offset:1536"
          :: "v"(lds), "v"(ga) : "memory");

      const unsigned long long oa = (unsigned long long)(uintptr_t)dst + (unsigned)(lane * 16);

      // Wait: this row's LDS fill complete AND prior row's LDS reads retired
      // before we read buf[r&1] / before anything can overwrite it.
      asm volatile("s_wait_asynccnt 0x0" ::: "memory");

      // LDS -> memory (ASYNCcnt). NT: 256 MB output is write-once, keep it
      // from thrashing L2 so the gather re-reads of x stay L2 hits.
      asm volatile(
          "global_store_async_from_lds_b128 %1, %0, off th:TH_STORE_NT\n\t"
          "global_store_async_from_lds_b128 %1, %0, off offset:512 th:TH_STORE_NT\n\t"
          "global_store_async_from_lds_b128 %1, %0, off offset:1024 th:TH_STORE_NT\n\t"
          "global_store_async_from_lds_b128 %1, %0, off offset:1536 th:TH_STORE_NT"
          :: "v"(lds), "v"(oa) : "memory");
    } else {
      // Masked frame: all-zero row via non-temporal b128 stores (STOREcnt).
      const v4f z = {0.f, 0.f, 0.f, 0.f};
      v4f* d4 = reinterpret_cast<v4f*>(dst);
      __builtin_nontemporal_store(z, d4 + lane);
      __builtin_nontemporal_store(z, d4 + lane + 32);
      __builtin_nontemporal_store(z, d4 + lane + 64);
      __builtin_nontemporal_store(z, d4 + lane + 96);
    }
  }
  // Drain async stores before the wave (and its LDS) goes away.
  asm volatile("s_wait_asynccnt 0x0" ::: "memory");
}

// ---------------------------------------------------------------------------
// Kernel 1: per-batch inclusive scan of durations (Hillis-Steele in LDS),
// write cum[] to workspace, mel_len to tail of d_out, init idx table to -1.
// ---------------------------------------------------------------------------
__global__ __launch_bounds__(T_) void lr_scan_kernel(const int* __restrict__ dur,
                                                     int* __restrict__ cum,
                                                     int* __restrict__ idxbuf,
                                                     float* __restrict__ mel_out) {
  __shared__ int s[T_];
  const int b = blockIdx.x;
  const int t = threadIdx.x;

  s[t] = dur[b * T_ + t];
  __syncthreads();
#pragma unroll
  for (int off = 1; off < T_; off <<= 1) {
    int v = (t >= off) ? s[t - off] : 0;
    __syncthreads();
    s[t] += v;
    __syncthreads();
  }
  cum[b * T_ + t] = s[t];
  if (t == T_ - 1) mel_out[b] = (float)s[T_ - 1];  // mel_len (exact in f32)

  // Sentinel -1 = "masked frame -> zeros"
#pragma unroll
  for (int f = t; f < ML_; f += T_) idxbuf[b * ML_ + f] = -1;
}

// ---------------------------------------------------------------------------
// Kernel 2: interval scatter. Thread t owns frames [cum[t-1], cum[t]) which
// map to source row t (== searchsorted(cum, f, side='right')). DUR_MAX=10 so
// each thread writes <= 9 entries. Frames >= min(mel_len,4096) keep -1.
// ---------------------------------------------------------------------------
__global__ __launch_bounds__(T_) void lr_scatter_kernel(const int* __restrict__ cum,
                                                        int* __restrict__ idxbuf) {
  const int b = blockIdx.x;
  const int t = threadIdx.x;
  const int c1 = cum[b * T_ + t];
  const int c0 = (t == 0) ? 0 : cum[b * T_ + t - 1];
  const int s  = (c0 < ML_) ? c0 : ML_;
  const int e  = (c1 < ML_) ? c1 : ML_;
  for (int f = s; f < e; ++f) idxbuf[b * ML_ + f] = t;
}

// ---------------------------------------------------------------------------
// Launcher. Inputs: d_in[0]=x (f32, B*T*D), d_in[1]=duration (i32, B*T),
// d_in[2]=max_len (ignored; constant 4096). d_out = out (B*ML*D f32) ++
// mel_len (B, emitted as f32). Workspace: idx table (B*ML i32) + cum (B*T i32)
// = 576 KB.
// ---------------------------------------------------------------------------
extern "C" void kernel_launch(void* const* d_in, const int* in_sizes, int n_in,
                              void* d_out, int out_size, void* d_ws, size_t ws_size,
                              hipStream_t stream) {
  (void)in_sizes; (void)n_in; (void)out_size; (void)ws_size;
  const float* x   = (const float*)d_in[0];
  const int*   dur = (const int*)d_in[1];
  float*       out = (float*)d_out;

  int* ws_idx = (int*)d_ws;                 // B_*ML_ ints (512 KB)
  int* ws_cum = ws_idx + (size_t)B_ * ML_;  // B_*T_  ints (64 KB)
  float* mel  = out + (size_t)B_ * ML_ * D_;

  lr_scan_kernel<<<B_, T_, 0, stream>>>(dur, ws_cum, ws_idx, mel);
  lr_scatter_kernel<<<B_, T_, 0, stream>>>(ws_cum, ws_idx);
  lr_expand_kernel<<<(B_ * ML_) / ROWS_PER_BLOCK, 32, 0, stream>>>(x, ws_idx, out);
}